// LstmCrfForNer_2662879724398
// MI455X (gfx1250) — compile-verified
//
#include <hip/hip_runtime.h>
#include <hip/hip_bf16.h>

// LSTM-CRF NER forward for MI455X (gfx1250, wave32).
// fp32 WMMA (V_WMMA_F32_16X16X4_F32) for all matmuls; W_hh LDS-stationary
// persistent recurrence kernel with per-timestep grid barrier.
// All global<->LDS staging vectorized to b128; next-slab prefetch in proj GEMM.

#define TT 256   // seq len
#define BB 32    // batch
#define EE 256   // embedding dim
#define HH 512   // hidden
#define G4 2048  // 4*H gates
#define X2 1024  // 2*H bidir concat
#define LL 9     // labels

#define WROW 516         // padded LDS row stride (floats); 2064B = 16B-multiple
#define NWG_TOTAL 64     // 32 col-slices * 2 directions in recurrence kernel

typedef __attribute__((ext_vector_type(2))) float v2f;
typedef __attribute__((ext_vector_type(8))) float v8f;

__device__ __forceinline__ float sigf(float x) { return 1.0f / (1.0f + __expf(-x)); }

// ---------------------------------------------------------------------------
// Embedding gather (float4): x0[t*B+b][e] = table[ids[b*T+t]][e]  (T,B,E)
// ---------------------------------------------------------------------------
__global__ void __launch_bounds__(256) embed_kernel(const int* __restrict__ ids,
                                                    const float* __restrict__ table,
                                                    float* __restrict__ x0) {
  int idx = blockIdx.x * 256 + threadIdx.x;  // < T*B*E/4
  int e4  = idx & 63;          // E/4 == 64
  int row = idx >> 6;          // t*B + b
  int t   = row >> 5;          // B == 32
  int b   = row & (BB - 1);
  int id  = ids[b * TT + t];
  const float4* src = reinterpret_cast<const float4*>(table + (size_t)id * EE);
  float4*       dst = reinterpret_cast<float4*>(x0 + (size_t)row * EE);
  dst[e4] = src[e4];
}

// ---------------------------------------------------------------------------
// Input projection GEMM: P[m][n] = sum_k X[m][k]*W[n][k] + bih[n] + bhh[n]
// M = T*B = 8192 rows, N = 2048 gates, K runtime (256 or 1024).
// Block tile 64x64, 8 waves, each wave: one 16x16 M-tile x two 16x16 N-tiles.
// blockIdx.z selects forward/reverse weight set.
// ---------------------------------------------------------------------------
__global__ void __launch_bounds__(256) proj_kernel(
    const float* __restrict__ X, int K,
    const float* __restrict__ Wf, const float* __restrict__ Wr,
    const float* __restrict__ bihf, const float* __restrict__ bhhf,
    const float* __restrict__ bihr, const float* __restrict__ bhhr,
    float* __restrict__ Pf, float* __restrict__ Pr) {
  const float *W, *bih, *bhh;
  float* P;
  if (blockIdx.z == 0) { W = Wf; bih = bihf; bhh = bhhf; P = Pf; }
  else                 { W = Wr; bih = bihr; bhh = bhhr; P = Pr; }

  __shared__ __align__(16) float sx[64 * 36];
  __shared__ __align__(16) float sw[64 * 36];

  const int tid  = threadIdx.x;
  const int lane = tid & 31;
  const int wave = tid >> 5;
  const int l16  = lane & 15;
  const int hiK  = (lane < 16) ? 0 : 2;   // A/B frag: low lanes K={0,1}, high lanes K={2,3}
  const int mT   = wave & 3;              // 0..3 -> M sub-tile
  const int nT0  = (wave >> 2) * 2;       // {0,2} -> pair of N sub-tiles

  const int m0 = blockIdx.y * 64;
  const int n0 = blockIdx.x * 64;

  v8f acc0 = {};
  v8f acc1 = {};

  // Per-thread staging coordinates: 64 rows x 8 float4 per matrix = 512 xfers,
  // 256 threads -> 2 float4 per thread per matrix per slab.
  const int sr0 = tid >> 2;               // 0..63 rows (thread covers rows sr0, with c4 pair)
  const int sc0 = (tid & 3) * 2;          // float4 col pair base: 0,2,4,6

  for (int kb = 0; kb < K; kb += 32) {
    __syncthreads();
    {
      const float4* xg = reinterpret_cast<const float4*>(X + (size_t)(m0 + sr0) * K + kb);
      const float4* wg = reinterpret_cast<const float4*>(W + (size_t)(n0 + sr0) * K + kb);
      float4* xs = reinterpret_cast<float4*>(sx + sr0 * 36);
      float4* wsld = reinterpret_cast<float4*>(sw + sr0 * 36);
      xs[sc0]     = xg[sc0];
      xs[sc0 + 1] = xg[sc0 + 1];
      wsld[sc0]     = wg[sc0];
      wsld[sc0 + 1] = wg[sc0 + 1];
      if (kb + 32 < K) {  // prefetch next K-slab (global_prefetch_b8)
        __builtin_prefetch(X + (size_t)(m0 + sr0) * K + kb + 32 + sc0 * 4, 0, 3);
        __builtin_prefetch(W + (size_t)(n0 + sr0) * K + kb + 32 + sc0 * 4, 0, 3);
      }
    }
    __syncthreads();
    for (int kk = 0; kk < 32; kk += 4) {
      int ak = kk + hiK;
      v2f a, b0, b1;
      a.x  = sx[(mT * 16 + l16) * 36 + ak];
      a.y  = sx[(mT * 16 + l16) * 36 + ak + 1];
      b0.x = sw[(nT0 * 16 + l16) * 36 + ak];
      b0.y = sw[(nT0 * 16 + l16) * 36 + ak + 1];
      b1.x = sw[((nT0 + 1) * 16 + l16) * 36 + ak];
      b1.y = sw[((nT0 + 1) * 16 + l16) * 36 + ak + 1];
      acc0 = __builtin_amdgcn_wmma_f32_16x16x4_f32(false, a, false, b0, (short)0, acc0, false, false);
      acc1 = __builtin_amdgcn_wmma_f32_16x16x4_f32(false, a, false, b1, (short)0, acc1, false, false);
    }
  }

  // Epilogue: C/D layout -> VGPR v holds M = v (lanes 0-15) or v+8 (lanes 16-31), N = lane%16
  for (int v = 0; v < 8; ++v) {
    int row = mT * 16 + ((lane < 16) ? v : v + 8);
    int na  = n0 + nT0 * 16 + l16;
    int nb  = n0 + (nT0 + 1) * 16 + l16;
    P[(size_t)(m0 + row) * G4 + na] = acc0[v] + bih[na] + bhh[na];
    P[(size_t)(m0 + row) * G4 + nb] = acc1[v] + bih[nb] + bhh[nb];
  }
}

// ---------------------------------------------------------------------------
// Persistent bidirectional LSTM recurrence for one layer.
// grid = (32, 2): 32 col-slices of 16 h-columns x 2 directions.
// Each WG keeps its 64x512 W_hh slice in LDS, loops over all T steps,
// grid-barriers between steps (counters zeroed before launch).
// Gate order (torch): i, f, g, o.  Output written to Xo (T,B,2H), dir offset.
// ---------------------------------------------------------------------------
__global__ void __launch_bounds__(256) lstm_rec_kernel(
    const float* __restrict__ Pf, const float* __restrict__ Pr,
    const float* __restrict__ Whf, const float* __restrict__ Whr,
    float* __restrict__ Xo, int* __restrict__ sync) {
  extern __shared__ float sm[];
  float* sW = sm;                               // 64 * WROW  (W_hh slice), 16B-aligned
  float* sH = sW + 64 * WROW;                   // 32 * WROW  (h_prev staging), 16B-aligned
  float* sG = sH + 32 * WROW;                   // 4 * 32 * 20 (gate tiles)
  float* sC = sG + 4 * 32 * 20;                 // 32 * 16    (cell state)

  const int dir    = blockIdx.y;
  const float* pre = dir ? Pr : Pf;
  const float* whh = dir ? Whr : Whf;
  const int c0     = blockIdx.x * 16;           // h-column base
  const int dirOff = dir * HH;

  const int tid  = threadIdx.x;
  const int lane = tid & 31;
  const int wave = tid >> 5;
  const int l16  = lane & 15;
  const int hiK  = (lane < 16) ? 0 : 2;
  const int mT   = wave >> 2;                   // 0..1 : batch rows 0-15 / 16-31
  const int nT   = wave & 3;                    // gate chunk: 0=i 1=f 2=g 3=o

  // Preload W_hh slice (float4): local row n -> gate col (n/16)*H + c0 + n%16
  for (int i = tid; i < 64 * (HH / 4); i += 256) {
    int n = i >> 7, kq = i & 127;               // HH/4 == 128
    int gcol = (n >> 4) * HH + c0 + (n & 15);
    reinterpret_cast<float4*>(sW + n * WROW)[kq] =
        reinterpret_cast<const float4*>(whh + (size_t)gcol * HH)[kq];
  }
  for (int i = tid; i < 32 * 16; i += 256) sC[i] = 0.0f;
  __syncthreads();

  for (int s = 0; s < TT; ++s) {
    const int tcur = dir ? (TT - 1 - s) : s;
    v8f acc = {};
    if (s > 0) {
      const int tprev = dir ? (tcur + 1) : (tcur - 1);
      for (int i = tid; i < 32 * (HH / 4); i += 256) {
        int b = i >> 7, kq = i & 127;
        reinterpret_cast<float4*>(sH + b * WROW)[kq] =
            reinterpret_cast<const float4*>(Xo + (size_t)(tprev * BB + b) * X2 + dirOff)[kq];
      }
      __syncthreads();
      for (int kk = 0; kk < HH; kk += 4) {
        int ak = kk + hiK;
        v2f a, bb;
        a.x  = sH[(mT * 16 + l16) * WROW + ak];
        a.y  = sH[(mT * 16 + l16) * WROW + ak + 1];
        bb.x = sW[(nT * 16 + l16) * WROW + ak];
        bb.y = sW[(nT * 16 + l16) * WROW + ak + 1];
        acc = __builtin_amdgcn_wmma_f32_16x16x4_f32(false, a, false, bb, (short)0, acc, false, false);
      }
    }

    // Gates (with precomputed input projection) into LDS
    for (int v = 0; v < 8; ++v) {
      int row = mT * 16 + ((lane < 16) ? v : v + 8);   // batch index
      float g = acc[v] + pre[(size_t)(tcur * BB + row) * G4 + nT * HH + c0 + l16];
      sG[(nT * 32 + row) * 20 + l16] = g;
    }
    __syncthreads();

    // Elementwise LSTM cell update + write h
    for (int e = tid; e < 32 * 16; e += 256) {
      int b = e >> 4, j = e & 15;
      float gi = sG[(0 * 32 + b) * 20 + j];
      float gf = sG[(1 * 32 + b) * 20 + j];
      float gg = sG[(2 * 32 + b) * 20 + j];
      float go = sG[(3 * 32 + b) * 20 + j];
      float c  = sigf(gf) * sC[e] + sigf(gi) * tanhf(gg);
      sC[e] = c;
      float h = sigf(go) * tanhf(c);
      Xo[(size_t)(tcur * BB + b) * X2 + dirOff + c0 + j] = h;
    }
    __threadfence();
    __syncthreads();

    if (s < TT - 1) {
      if (tid == 0) {
        atomicAdd(&sync[s], 1);
        volatile int* vp = sync + s;
        while (*vp < NWG_TOTAL) { __builtin_amdgcn_s_sleep(1); }
      }
      __syncthreads();
      __threadfence();
    }
  }
}

// ---------------------------------------------------------------------------
// Emissions: em[(b*T+t)*L + l] = dot(x2[t*B+b], cls_w[l]) + cls_b[l]  (float4 dot)
// ---------------------------------------------------------------------------
__global__ void __launch_bounds__(256) emis_kernel(const float* __restrict__ X,
                                                   const float* __restrict__ Wc,
                                                   const float* __restrict__ bc,
                                                   float* __restrict__ em) {
  int i = blockIdx.x * 256 + threadIdx.x;
  if (i >= BB * TT * LL) return;
  int l  = i % LL;
  int bt = i / LL;
  int b  = bt >> 8;   // / T (T == 256)
  int t  = bt & 255;
  const float4* xr = reinterpret_cast<const float4*>(X + (size_t)(t * BB + b) * X2);
  const float4* wr = reinterpret_cast<const float4*>(Wc + (size_t)l * X2);
  float sum = 0.f;
  for (int k = 0; k < X2 / 4; ++k) {
    float4 xv = xr[k], wv = wr[k];
    sum = fmaf(xv.x, wv.x, sum);
    sum = fmaf(xv.y, wv.y, sum);
    sum = fmaf(xv.z, wv.z, sum);
    sum = fmaf(xv.w, wv.w, sum);
  }
  em[i] = sum + bc[l];
}

// ---------------------------------------------------------------------------
// CRF NLL (single block, 288 = 32*9 threads; thread = (b, j))
// ---------------------------------------------------------------------------
__global__ void __launch_bounds__(288) crf_kernel(const float* __restrict__ em,
                                                  const int* __restrict__ labels,
                                                  const int* __restrict__ mask,
                                                  const float* __restrict__ stw,
                                                  const float* __restrict__ enw,
                                                  const float* __restrict__ trans,
                                                  float* __restrict__ out) {
  __shared__ float alpha[BB][LL];
  __shared__ float score[BB];
  __shared__ int   prev[BB];
  __shared__ float diff[BB];

  const int tid = threadIdx.x;     // 0..287
  const int b   = tid / LL;
  const int j   = tid % LL;

  {
    int tg = labels[b * TT + 0]; if (tg == -100) tg = 0;
    alpha[b][j] = stw[j] + em[(b * TT + 0) * LL + j];
    if (j == 0) { score[b] = stw[tg] + em[(b * TT + 0) * LL + tg]; prev[b] = tg; }
  }
  __syncthreads();

  for (int t = 1; t < TT; ++t) {
    float mk = (float)mask[b * TT + t];
    float mx = -3.4e38f;
    for (int i = 0; i < LL; ++i)
      mx = fmaxf(mx, alpha[b][i] + trans[i * LL + j]);
    float se = 0.f;
    for (int i = 0; i < LL; ++i)
      se += __expf(alpha[b][i] + trans[i * LL + j] - mx);
    float nxt = mx + __logf(se) + em[(b * TT + t) * LL + j];
    int tg = labels[b * TT + t]; if (tg == -100) tg = 0;
    __syncthreads();
    if (mk > 0.f) alpha[b][j] = nxt;
    if (j == 0) {
      score[b] += (trans[prev[b] * LL + tg] + em[(b * TT + t) * LL + tg]) * mk;
      if (mk > 0.f) prev[b] = tg;
    }
    __syncthreads();
  }

  if (j == 0) {
    float sc = score[b] + enw[prev[b]];
    float mx = -3.4e38f;
    for (int jj = 0; jj < LL; ++jj) mx = fmaxf(mx, alpha[b][jj] + enw[jj]);
    float se = 0.f;
    for (int jj = 0; jj < LL; ++jj) se += __expf(alpha[b][jj] + enw[jj] - mx);
    diff[b] = sc - (mx + __logf(se));
  }
  __syncthreads();
  if (tid == 0) {
    float s = 0.f;
    for (int k = 0; k < BB; ++k) s += diff[k];
    out[0] = -s / (float)BB;
  }
}

// ---------------------------------------------------------------------------
extern "C" void kernel_launch(void* const* d_in, const int* in_sizes, int n_in,
                              void* d_out, int out_size, void* d_ws, size_t ws_size,
                              hipStream_t stream) {
  (void)in_sizes; (void)n_in; (void)out_size; (void)ws_size;

  const int*   ids    = (const int*)  d_in[0];
  const int*   amask  = (const int*)  d_in[1];
  const int*   labels = (const int*)  d_in[2];
  const float* emb    = (const float*)d_in[3];
  const float* wih0f = (const float*)d_in[4];
  const float* whh0f = (const float*)d_in[5];
  const float* bih0f = (const float*)d_in[6];
  const float* bhh0f = (const float*)d_in[7];
  const float* wih0r = (const float*)d_in[8];
  const float* whh0r = (const float*)d_in[9];
  const float* bih0r = (const float*)d_in[10];
  const float* bhh0r = (const float*)d_in[11];
  const float* wih1f = (const float*)d_in[12];
  const float* whh1f = (const float*)d_in[13];
  const float* bih1f = (const float*)d_in[14];
  const float* bhh1f = (const float*)d_in[15];
  const float* wih1r = (const float*)d_in[16];
  const float* whh1r = (const float*)d_in[17];
  const float* bih1r = (const float*)d_in[18];
  const float* bhh1r = (const float*)d_in[19];
  const float* clsw  = (const float*)d_in[20];
  const float* clsb  = (const float*)d_in[21];
  const float* cstart = (const float*)d_in[22];
  const float* cend   = (const float*)d_in[23];
  const float* ctrans = (const float*)d_in[24];
  float* out = (float*)d_out;

  // Workspace layout
  int*   sync = (int*)d_ws;                        // 256 counters (zeroed per recurrence)
  float* ws   = (float*)d_ws;
  float* x0 = ws + 1024;                           // (T,B,E)
  float* x1 = x0 + (size_t)TT * BB * EE;           // (T,B,2H) layer0 out
  float* x2 = x1 + (size_t)TT * BB * X2;           // (T,B,2H) layer1 out
  float* pf = x2 + (size_t)TT * BB * X2;           // (T,B,4H) pre fwd
  float* pr = pf + (size_t)TT * BB * G4;           // (T,B,4H) pre rev

  const size_t smem = (size_t)(64 * WROW + 32 * WROW + 4 * 32 * 20 + 32 * 16) * sizeof(float);

  embed_kernel<<<dim3((TT * BB * EE / 4) / 256), 256, 0, stream>>>(ids, emb, x0);

  proj_kernel<<<dim3(G4 / 64, (TT * BB) / 64, 2), 256, 0, stream>>>(
      x0, EE, wih0f, wih0r, bih0f, bhh0f, bih0r, bhh0r, pf, pr);

  hipMemsetAsync(sync, 0, TT * sizeof(int), stream);
  lstm_rec_kernel<<<dim3(HH / 16, 2), 256, smem, stream>>>(pf, pr, whh0f, whh0r, x1, sync);

  proj_kernel<<<dim3(G4 / 64, (TT * BB) / 64, 2), 256, 0, stream>>>(
      x1, X2, wih1f, wih1r, bih1f, bhh1f, bih1r, bhh1r, pf, pr);

  hipMemsetAsync(sync, 0, TT * sizeof(int), stream);
  lstm_rec_kernel<<<dim3(HH / 16, 2), 256, smem, stream>>>(pf, pr, whh1f, whh1r, x2, sync);

  emis_kernel<<<dim3((BB * TT * LL + 255) / 256), 256, 0, stream>>>(x2, clsw, clsb, out + 1);

  crf_kernel<<<1, 288, 0, stream>>>(out + 1, labels, amask, cstart, cend, ctrans, out);
}